// PsiU_52072183497111
// MI455X (gfx1250) — compile-verified
//
#include <hip/hip_runtime.h>
#include <hip/hip_bf16.h>
#include <math.h>
#include <stdint.h>

// ---------------------------------------------------------------------------
// Problem constants
// ---------------------------------------------------------------------------
#define DIM_IN   64
#define DIM_OUT  64
#define DIM_XI   128
#define LDIM     96
#define NBATCH   131072
#define NBIG     352            // 2*DIM_XI + LDIM
#define EPSV     0.001f

// ---------------------------------------------------------------------------
// Workspace layout (float offsets)
// ---------------------------------------------------------------------------
// All weight matrices stored transposed (K x N, row-major) = WMMA B-operand
// layout so B-fragment loads are coalesced (16 consecutive lanes -> 16
// consecutive floats of a row).
#define OFF_W1    0          /* D12^T : 64 x 96   (u -> base)   */
#define OFF_W6    6144       /* D22^T : 64 x 64   (u -> y)      */
#define OFF_W5    10240      /* D21^T : 96 x 64   (w -> y)      */
#define OFF_W3    16384      /* B2^T  : 64 x 128  (u -> inner)  */
#define OFF_W2    24576      /* B1^T  : 96 x 128  (w -> inner)  */
#define OFF_D11   36864      /* D11   : 96 x 96  row-major (scan VALU part) */
#define OFF_D11T  46080      /* D11^T : 96 x 96  (scan WMMA part, B layout) */
#define OFF_LAMI  55296      /* 1/Lam : 96 */
#define OFF_XB    55392      /* xi @ C1^T : 96  */
#define OFF_XF    55488      /* xi @ Fm^T : 128 */
#define OFF_E     55616      /* E : 128 x 128 (destroyed by GJ) */
#define OFF_IE    72000      /* inv(E) : 128 x 128 */
#define OFF_P     88384      /* P = inv(E)^T C2^T : 128 x 64 (inner -> y) */
#define WS_FLOATS 96576
#define PREP1_N   72000      /* elements filled by prep1 (through E) */

typedef float v2f __attribute__((ext_vector_type(2)));
typedef float v8f __attribute__((ext_vector_type(8)));

// ---------------------------------------------------------------------------
// H(a,b) = (X^T X)[a][b] + EPS*(a==b), computed on the fly (X is 352x352)
// ---------------------------------------------------------------------------
__device__ __forceinline__ float Hdot(const float* __restrict__ X, int a, int b) {
    float s = 0.f;
    for (int r = 0; r < NBIG; ++r)
        s = fmaf(X[r * NBIG + a], X[r * NBIG + b], s);
    if (a == b) s += EPSV;
    return s;
}

// ---------------------------------------------------------------------------
// prep1: derived weights (one element per thread)
// ---------------------------------------------------------------------------
__global__ void prep1_kernel(const float* __restrict__ X,
                             const float* __restrict__ Y,
                             const float* __restrict__ xi,
                             const float* __restrict__ B2,
                             const float* __restrict__ D21,
                             const float* __restrict__ D22,
                             const float* __restrict__ D12,
                             float* __restrict__ ws) {
    int i = blockIdx.x * blockDim.x + threadIdx.x;

    if (i < 6144) {                       // W1 = D12^T (64x96); D12 is 96x64
        int k = i / 96, n = i % 96;
        ws[OFF_W1 + i] = D12[n * 64 + k];
        return;
    }
    i -= 6144;
    if (i < 4096) {                       // W6 = D22^T (64x64); D22 is 64x64
        int k = i / 64, n = i % 64;
        ws[OFF_W6 + i] = D22[n * 64 + k];
        return;
    }
    i -= 4096;
    if (i < 6144) {                       // W5 = D21^T (96x64); D21 is 64x96
        int k = i / 64, n = i % 64;
        ws[OFF_W5 + i] = D21[n * 96 + k];
        return;
    }
    i -= 6144;
    if (i < 8192) {                       // W3 = B2^T (64x128); B2 is 128x64
        int k = i / 128, n = i % 128;
        ws[OFF_W3 + i] = B2[n * 64 + k];
        return;
    }
    i -= 8192;
    if (i < 12288) {                      // W2 = B1^T (96x128); B1[n][k] = H[224+n][128+k]
        int k = i / 128, n = i % 128;
        ws[OFF_W2 + i] = Hdot(X, 224 + n, 128 + k);
        return;
    }
    i -= 12288;
    if (i < 9216) {                       // D11[r][c] = -(r>c) * H22[r][c]
        int r = i / 96, c = i % 96;
        ws[OFF_D11 + i] = (r > c) ? -Hdot(X, 128 + r, 128 + c) : 0.f;
        return;
    }
    i -= 9216;
    if (i < 9216) {                       // D11T[j][r] = D11[r][j]
        int j = i / 96, r = i % 96;
        ws[OFF_D11T + i] = (r > j) ? -Hdot(X, 128 + r, 128 + j) : 0.f;
        return;
    }
    i -= 9216;
    if (i < 96) {                         // 1/Lam = 2 / H22[i][i]
        ws[OFF_LAMI + i] = 2.f / Hdot(X, 128 + i, 128 + i);
        return;
    }
    i -= 96;
    if (i < 96) {                         // xibase[i] = (xi @ C1^T)[i] = -sum_b xi[b] H[128+i][b]
        float s = 0.f;
        for (int b = 0; b < 128; ++b) s = fmaf(xi[b], Hdot(X, 128 + i, b), s);
        ws[OFF_XB + i] = -s;
        return;
    }
    i -= 96;
    if (i < 128) {                        // xiF[i] = (xi @ Fm^T)[i] = sum_b xi[b] H[224+i][b]
        float s = 0.f;
        for (int b = 0; b < 128; ++b) s = fmaf(xi[b], Hdot(X, 224 + i, b), s);
        ws[OFF_XF + i] = s;
        return;
    }
    i -= 128;
    if (i < 16384) {                      // E = 0.5*(H11 + H33 + Y - Y^T)
        int a = i / 128, b = i % 128;
        ws[OFF_E + i] = 0.5f * (Hdot(X, a, b) + Hdot(X, 224 + a, 224 + b)
                                + Y[a * 128 + b] - Y[b * 128 + a]);
        return;
    }
}

// ---------------------------------------------------------------------------
// prep2: Gauss-Jordan inverse of E (128x128, one workgroup, thread t owns
// column t), then P = inv(E)^T @ C2^T  (128 x 64)
// ---------------------------------------------------------------------------
__global__ __launch_bounds__(128) void prep2_kernel(const float* __restrict__ C2,
                                                    float* __restrict__ ws) {
    float* E  = ws + OFF_E;
    float* IE = ws + OFF_IE;
    float* P  = ws + OFF_P;
    const int t = threadIdx.x;

    for (int r = 0; r < 128; ++r) IE[r * 128 + t] = (r == t) ? 1.f : 0.f;
    __shared__ float colk[128];

    for (int k = 0; k < 128; ++k) {
        __syncthreads();
        float pvi = 1.f / E[k * 128 + k];
        E[k * 128 + t]  *= pvi;
        IE[k * 128 + t] *= pvi;
        __syncthreads();
        colk[t] = E[t * 128 + k];
        __syncthreads();
        float Ekt  = E[k * 128 + t];
        float IEkt = IE[k * 128 + t];
        for (int r = 0; r < 128; ++r) {
            if (r == k) continue;
            float f = colk[r];
            E[r * 128 + t]  = fmaf(-f, Ekt,  E[r * 128 + t]);
            IE[r * 128 + t] = fmaf(-f, IEkt, IE[r * 128 + t]);
        }
    }
    __syncthreads();
    // P[k][n] = sum_j inv(E)[j][k] * C2[n][j]   (C2 is 64 x 128)
    for (int i = t; i < 128 * 64; i += 128) {
        int k = i >> 6, n = i & 63;
        float s = 0.f;
        for (int j = 0; j < 128; ++j)
            s = fmaf(IE[j * 128 + k], C2[n * 128 + j], s);
        P[i] = s;
    }
}

// ---------------------------------------------------------------------------
// WMMA helpers: D(16x16,f32) += A(16xK, LDS row-major) * B(KxN, global
// row-major), via V_WMMA_F32_16X16X4_F32.
//
// Layouts per CDNA5 ISA 7.12.2:
//  A 16x4 f32 : lane m=lane&15; VGPR0 = A[m][k0 + 2*hi], VGPR1 = +1
//  B 4x16 f32 : lane n=lane&15; VGPR0 = B[k0 + 2*hi][n],  VGPR1 = +1
//  C/D 16x16  : VGPR v -> row (v + 8*hi), col = lane&15
// ---------------------------------------------------------------------------
__device__ __forceinline__ v8f wmma_acc(const float* __restrict__ A, int lda,
                                        const float* __restrict__ Bw, int ldb,
                                        int n0, int K, v8f c, int lane) {
    const int m  = lane & 15;
    const int hi = lane >> 4;
    const int n  = n0 + m;
    for (int k = 0; k < K; k += 4) {
        const int kb = k + 2 * hi;
        v2f a; a.x = A[m * lda + kb];       a.y = A[m * lda + kb + 1];
        v2f b; b.x = Bw[kb * ldb + n];      b.y = Bw[(kb + 1) * ldb + n];
        c = __builtin_amdgcn_wmma_f32_16x16x4_f32(false, a, false, b,
                                                  (short)0, c, false, false);
    }
    return c;
}

__device__ __forceinline__ void store_d_lds(float* __restrict__ Wd, int ldw,
                                            int n0, v8f c, int lane) {
    const int n  = n0 + (lane & 15);
    const int hi = lane >> 4;
#pragma unroll
    for (int v = 0; v < 8; ++v) Wd[(v + 8 * hi) * ldw + n] = c[v];
}

__device__ __forceinline__ v8f load_c_lds(const float* __restrict__ Wd, int ldw,
                                          int n0, int lane) {
    const int n  = n0 + (lane & 15);
    const int hi = lane >> 4;
    v8f c;
#pragma unroll
    for (int v = 0; v < 8; ++v) c[v] = Wd[(v + 8 * hi) * ldw + n];
    return c;
}

#define WAIT_DS()    asm volatile("s_wait_dscnt 0" ::: "memory")
#define WAIT_ASYNC() asm volatile("s_wait_asynccnt 0" ::: "memory")

// Async global->LDS B128 copy (CDNA5 TDM-adjacent path, ASYNCcnt-tracked).
// the immediate offset is added to both the LDS and the global address
#define ASYNC_LD_B128(ldsv, gaddr, imm)                                        \
    asm volatile("global_load_async_to_lds_b128 %0, %1, off offset:" #imm     \
                 :: "v"(ldsv), "v"(gaddr) : "memory")

// ---------------------------------------------------------------------------
// fwd: 4 waves/block, each wave owns a private 16-row batch stripe.
// ---------------------------------------------------------------------------
__global__ __launch_bounds__(128) void fwd_kernel(const float* __restrict__ u,
                                                  const float* __restrict__ ws,
                                                  float* __restrict__ out) {
    __shared__ __align__(16) float sU[4][16 * 64];   // u stripe
    __shared__ __align__(16) float sW[4][16 * 96];   // base -> w (scan in place)
    __shared__ __align__(16) float sC[4][16 * 16];   // inner chunk scratch

    const int tid  = threadIdx.x;
    const int wv   = tid >> 5;
    const int lane = tid & 31;
    const int m    = lane & 15;
    const int hi   = lane >> 4;
    const int r0   = blockIdx.x * 64 + wv * 16;     // batch-row base of stripe

    float* myU = &sU[wv][0];
    float* myW = &sW[wv][0];
    float* myC = &sC[wv][0];

    // ---- async-copy u stripe (16 x 64 f32 = 4KB, contiguous) into LDS -------
    {
        uint32_t ldsa = (uint32_t)(uintptr_t)myU + (uint32_t)lane * 16u;
        uint64_t ga   = (uint64_t)(uintptr_t)u + (uint64_t)r0 * 256u
                        + (uint64_t)lane * 16u;
        ASYNC_LD_B128(ldsa, ga, 0);
        ASYNC_LD_B128(ldsa, ga, 512);
        ASYNC_LD_B128(ldsa, ga, 1024);
        ASYNC_LD_B128(ldsa, ga, 1536);
        ASYNC_LD_B128(ldsa, ga, 2048);
        ASYNC_LD_B128(ldsa, ga, 2560);
        ASYNC_LD_B128(ldsa, ga, 3072);
        ASYNC_LD_B128(ldsa, ga, 3584);
    }
    // warm the weight path into WGP$/L2 while the async copy is in flight
    __builtin_prefetch(ws + OFF_W1, 0, 1);
    __builtin_prefetch(ws + OFF_XB, 0, 1);
    WAIT_ASYNC();

    // ---- base = xi@C1^T + u @ D12^T  ->  sW (16 x 96) -----------------------
    const float* xb = ws + OFF_XB;
    for (int nt = 0; nt < 6; ++nt) {
        float b0 = xb[nt * 16 + m];
        v8f c = {b0, b0, b0, b0, b0, b0, b0, b0};
        c = wmma_acc(myU, 64, ws + OFF_W1, 96, nt * 16, 64, c, lane);
        store_d_lds(myW, 96, nt * 16, c, lane);
    }
    WAIT_DS();

    // ---- sequential tanh scan, blocked by 16 --------------------------------
    // Inter-block contribution via WMMA against D11^T; intra-block triangle
    // on VALU, split even/odd j across half-waves, combined with shfl_xor.
    {
        const float* d11  = ws + OFF_D11;
        const float* lami = ws + OFF_LAMI;
        float* Wrow = myW + m * 96;
        for (int blk = 0; blk < 6; ++blk) {
            const int i0 = blk * 16;
            if (blk) {
                v8f c = load_c_lds(myW, 96, i0, lane);
                c = wmma_acc(myW, 96, ws + OFF_D11T, 96, i0, i0, c, lane);
                store_d_lds(myW, 96, i0, c, lane);
                WAIT_DS();
            }
            for (int i = i0; i < i0 + 16; ++i) {
                float s = 0.f;
                for (int j = i0 + hi; j < i; j += 2)
                    s = fmaf(Wrow[j], d11[i * 96 + j], s);
                s += __shfl_xor(s, 16);
                float v  = Wrow[i] + s;
                float tv = tanhf(v * lami[i]);
                if (hi == 0) Wrow[i] = tv;
                WAIT_DS();
            }
        }
    }

    // ---- y = u@D22^T + w@D21^T + inner@P  (inner in 16-col chunks) ----------
    v8f yacc[4];
    {
        v8f z = {0.f, 0.f, 0.f, 0.f, 0.f, 0.f, 0.f, 0.f};
#pragma unroll
        for (int nt = 0; nt < 4; ++nt) {
            v8f c = z;
            c = wmma_acc(myU, 64, ws + OFF_W6, 64, nt * 16, 64, c, lane);
            c = wmma_acc(myW, 96, ws + OFF_W5, 64, nt * 16, 96, c, lane);
            yacc[nt] = c;
        }
    }
    const float* xf = ws + OFF_XF;
    for (int kt = 0; kt < 8; ++kt) {
        // inner chunk: cols kt*16 .. kt*16+15 of (xi@Fm^T + w@B1^T + u@B2^T)
        float f0 = xf[kt * 16 + m];
        v8f ci = {f0, f0, f0, f0, f0, f0, f0, f0};
        ci = wmma_acc(myU, 64, ws + OFF_W3, 128, kt * 16, 64, ci, lane);
        ci = wmma_acc(myW, 96, ws + OFF_W2, 128, kt * 16, 96, ci, lane);
        store_d_lds(myC, 16, 0, ci, lane);
        WAIT_DS();
#pragma unroll
        for (int nt = 0; nt < 4; ++nt)
            yacc[nt] = wmma_acc(myC, 16, ws + OFF_P + kt * 16 * 64, 64,
                                nt * 16, 16, yacc[nt], lane);
    }

    // ---- store y (16 x 64) --------------------------------------------------
#pragma unroll
    for (int nt = 0; nt < 4; ++nt) {
#pragma unroll
        for (int v = 0; v < 8; ++v)
            out[(r0 + v + 8 * hi) * 64 + nt * 16 + m] = yacc[nt][v];
    }
}

// ---------------------------------------------------------------------------
// Launch
// ---------------------------------------------------------------------------
extern "C" void kernel_launch(void* const* d_in, const int* in_sizes, int n_in,
                              void* d_out, int out_size, void* d_ws, size_t ws_size,
                              hipStream_t stream) {
    const float* u   = (const float*)d_in[0];   // (B,1,64)
    const float* xi  = (const float*)d_in[1];   // (1,128)
    const float* X   = (const float*)d_in[2];   // 352*352
    const float* Y   = (const float*)d_in[3];   // 128*128
    const float* B2  = (const float*)d_in[4];   // 128*64
    const float* C2  = (const float*)d_in[5];   // 64*128
    const float* D21 = (const float*)d_in[6];   // 64*96
    const float* D22 = (const float*)d_in[7];   // 64*64
    const float* D12 = (const float*)d_in[8];   // 96*64
    float* out = (float*)d_out;
    float* ws  = (float*)d_ws;

    prep1_kernel<<<(PREP1_N + 255) / 256, 256, 0, stream>>>(X, Y, xi, B2, D21,
                                                            D22, D12, ws);
    prep2_kernel<<<1, 128, 0, stream>>>(C2, ws);
    fwd_kernel<<<NBATCH / 64, 128, 0, stream>>>(u, ws, out);
}